// ADDNODE_GNN_52097953300917
// MI455X (gfx1250) — compile-verified
//
#include <hip/hip_runtime.h>
#include <hip/hip_bf16.h>

// ---------------------------------------------------------------------------
// ADDNODE_GNN on MI455X (gfx1250): f32 WMMA 16x16x4 pipeline.
//   Kernel A: node MLP (256->256 relu ->128) + L2 row normalize  -> mvc in d_ws
//   Kernel B: edge gather + (256->64 relu) GEMM via WMMA (2 M-tiles/wave to
//             amortize W_fc1 B-fragments) + 64->2 head + gumbel softmax
//             straight-through -> train_mask (3*E floats)
// ---------------------------------------------------------------------------

typedef float v2f __attribute__((ext_vector_type(2)));
typedef float v8f __attribute__((ext_vector_type(8)));

#define NNODES      100000
#define KDIM        256          // L*D = TRACE_HID input K
#define MVCD        128
#define NEDGE       1600000
#define EIDX_STRIDE 4800000      // edge_index row stride = 3*NUM_EDGE
#define TAU_INV     2.0f         // 1/0.5

#define WMMA_F32(A, B, C) \
    __builtin_amdgcn_wmma_f32_16x16x4_f32(false, (A), false, (B), (short)0, (C), false, false)

// ------------------------------ Kernel A -----------------------------------
// grid = N/16 = 6250 blocks, 256 threads (8 waves). One 16-node tile / block.
__global__ __launch_bounds__(256) void node_mlp_kernel(
    const float* __restrict__ trace,   // (2, N, 128)
    const float* __restrict__ Wlin,    // (256, 256) row-major
    const float* __restrict__ Wlin2,   // (128, 256) row-major
    float* __restrict__ mvc)           // (N, 128) normalized output
{
    __shared__ float xs[16 * 260];     // x tile, padded stride 260
    __shared__ float hs[16 * 260];     // relu(x@Wlin^T) tile
    __shared__ float ms[16 * 132];     // mvc tile pre-norm
    __shared__ float scale[16];

    const int tid  = threadIdx.x;
    const int wave = tid >> 5;
    const int lane = tid & 31;
    const int m0   = blockIdx.x * 16;

    // Stage x(16x256): x[r][k] = trace[k<128?0:1][m0+r][k%128]
    const float* t0 = trace;
    const float* t1 = trace + (size_t)NNODES * 128;
    {
        const int k = tid;             // 0..255
        #pragma unroll
        for (int r = 0; r < 16; ++r) {
            const int node = m0 + r;
            float v = (k < 128) ? t0[(size_t)node * 128 + k]
                                : t1[(size_t)node * 128 + (k - 128)];
            xs[r * 260 + k] = v;
        }
    }
    __syncthreads();

    const int half = lane >> 4;        // 0: lanes 0-15, 1: lanes 16-31
    const int mrow = lane & 15;        // A-matrix row / B-matrix col
    const int kk   = half * 2;         // K sub-offset within a 4-wide step

    // ---- GEMM1: h = relu(x @ Wlin^T). Wave w covers output cols [32w,32w+32)
    v8f acc0 = {0.f,0.f,0.f,0.f,0.f,0.f,0.f,0.f};
    v8f acc1 = {0.f,0.f,0.f,0.f,0.f,0.f,0.f,0.f};
    const int n0a = wave * 32;
    const int n0b = n0a + 16;
    const float* wra = Wlin + (size_t)(n0a + mrow) * KDIM; // B[k][n]=Wlin[n][k]
    const float* wrb = Wlin + (size_t)(n0b + mrow) * KDIM;

    #pragma unroll 4
    for (int k0 = 0; k0 < KDIM; k0 += 4) {
        v2f a  = *(const v2f*)&xs[mrow * 260 + k0 + kk];
        v2f b0 = *(const v2f*)&wra[k0 + kk];
        v2f b1 = *(const v2f*)&wrb[k0 + kk];
        acc0 = WMMA_F32(a, b0, acc0);
        acc1 = WMMA_F32(a, b1, acc1);
    }
    #pragma unroll
    for (int r = 0; r < 8; ++r) {      // C/D layout: row = r + 8*half, col=n0+mrow
        const int row = r + half * 8;
        hs[row * 260 + n0a + mrow] = fmaxf(acc0[r], 0.f);
        hs[row * 260 + n0b + mrow] = fmaxf(acc1[r], 0.f);
    }
    __syncthreads();

    // ---- GEMM2: mvc = h @ Wlin2^T. Wave w covers cols [16w,16w+16)
    v8f macc = {0.f,0.f,0.f,0.f,0.f,0.f,0.f,0.f};
    const int n0 = wave * 16;
    const float* wr2 = Wlin2 + (size_t)(n0 + mrow) * KDIM;
    #pragma unroll 4
    for (int k0 = 0; k0 < KDIM; k0 += 4) {
        v2f a = *(const v2f*)&hs[mrow * 260 + k0 + kk];
        v2f b = *(const v2f*)&wr2[k0 + kk];
        macc = WMMA_F32(a, b, macc);
    }
    #pragma unroll
    for (int r = 0; r < 8; ++r) {
        const int row = r + half * 8;
        ms[row * 132 + n0 + mrow] = macc[r];
    }
    __syncthreads();

    if (tid < 16) {                    // row L2 norms
        float s = 0.f;
        #pragma unroll 8
        for (int c = 0; c < MVCD; ++c) {
            float v = ms[tid * 132 + c];
            s += v * v;
        }
        s = fmaxf(sqrtf(s), 1e-12f);
        scale[tid] = 1.0f / s;
    }
    __syncthreads();

    #pragma unroll
    for (int i = 0; i < 8; ++i) {      // 2048 normalized elems, coalesced
        const int idx = tid + i * 256;
        const int row = idx >> 7;
        const int col = idx & 127;
        mvc[(size_t)(m0 + row) * MVCD + col] = ms[row * 132 + col] * scale[row];
    }
}

// ------------------------------ Kernel B -----------------------------------
// grid = E/256 = 6250 blocks, 256 threads (8 waves). 32 edges per wave
// (two 16-row M-tiles sharing the W_fc1 B-fragments each K-step).
__global__ __launch_bounds__(256) void edge_mask_kernel(
    const float* __restrict__ mvc,     // (N,128) normalized, L2-resident
    const float* __restrict__ Wfc1,    // (64,256)
    const float* __restrict__ bfc1,    // (64)
    const float* __restrict__ Wfc2,    // (2,64)
    const float* __restrict__ bfc2,    // (2)
    const float* __restrict__ gumbel,  // (E,2)
    const int*   __restrict__ eidx,    // (2, 3E)
    float* __restrict__ out)           // (3E)
{
    __shared__ float h2s[128 * 68];    // relu hidden for one tile set (pad 68)

    const int tid  = threadIdx.x;
    const int wave = tid >> 5;
    const int lane = tid & 31;
    const int half = lane >> 4;
    const int mrow = lane & 15;
    const int kk   = half * 2;

    const long eBase = (long)blockIdx.x * 256;
    const long eA    = eBase + wave * 32 + mrow;        // M-tile A row edge
    const long eB    = eA + 16;                         // M-tile B row edge

    const int srcA = eidx[eA];
    const int dstA = eidx[(long)EIDX_STRIDE + eA];
    const int srcB = eidx[eB];
    const int dstB = eidx[(long)EIDX_STRIDE + eB];
    const float* psA = mvc + (size_t)srcA * MVCD;
    const float* pdA = mvc + (size_t)dstA * MVCD;
    const float* psB = mvc + (size_t)srcB * MVCD;
    const float* pdB = mvc + (size_t)dstB * MVCD;

    v8f accA[4], accB[4];
    #pragma unroll
    for (int t = 0; t < 4; ++t) {
        accA[t] = (v8f){0.f,0.f,0.f,0.f,0.f,0.f,0.f,0.f};
        accB[t] = (v8f){0.f,0.f,0.f,0.f,0.f,0.f,0.f,0.f};
    }

    const float* wb0 = Wfc1 + (size_t)( 0 + mrow) * KDIM;  // B[k][n]=Wfc1[n][k]
    const float* wb1 = Wfc1 + (size_t)(16 + mrow) * KDIM;
    const float* wb2 = Wfc1 + (size_t)(32 + mrow) * KDIM;
    const float* wb3 = Wfc1 + (size_t)(48 + mrow) * KDIM;

    // K in [0,128): src half of comb
    #pragma unroll 2
    for (int k0 = 0; k0 < 128; k0 += 4) {
        v2f aA = *(const v2f*)&psA[k0 + kk];
        v2f aB = *(const v2f*)&psB[k0 + kk];
        v2f b0 = *(const v2f*)&wb0[k0 + kk];
        v2f b1 = *(const v2f*)&wb1[k0 + kk];
        v2f b2 = *(const v2f*)&wb2[k0 + kk];
        v2f b3 = *(const v2f*)&wb3[k0 + kk];
        accA[0] = WMMA_F32(aA, b0, accA[0]);
        accB[0] = WMMA_F32(aB, b0, accB[0]);
        accA[1] = WMMA_F32(aA, b1, accA[1]);
        accB[1] = WMMA_F32(aB, b1, accB[1]);
        accA[2] = WMMA_F32(aA, b2, accA[2]);
        accB[2] = WMMA_F32(aB, b2, accB[2]);
        accA[3] = WMMA_F32(aA, b3, accA[3]);
        accB[3] = WMMA_F32(aB, b3, accB[3]);
    }
    // K in [128,256): dst half of comb
    #pragma unroll 2
    for (int k0 = 0; k0 < 128; k0 += 4) {
        v2f aA = *(const v2f*)&pdA[k0 + kk];
        v2f aB = *(const v2f*)&pdB[k0 + kk];
        v2f b0 = *(const v2f*)&wb0[128 + k0 + kk];
        v2f b1 = *(const v2f*)&wb1[128 + k0 + kk];
        v2f b2 = *(const v2f*)&wb2[128 + k0 + kk];
        v2f b3 = *(const v2f*)&wb3[128 + k0 + kk];
        accA[0] = WMMA_F32(aA, b0, accA[0]);
        accB[0] = WMMA_F32(aB, b0, accB[0]);
        accA[1] = WMMA_F32(aA, b1, accA[1]);
        accB[1] = WMMA_F32(aB, b1, accB[1]);
        accA[2] = WMMA_F32(aA, b2, accA[2]);
        accB[2] = WMMA_F32(aB, b2, accB[2]);
        accA[3] = WMMA_F32(aA, b3, accA[3]);
        accB[3] = WMMA_F32(aB, b3, accB[3]);
    }

    // ---- two passes over the shared h2 buffer: tile A then tile B ---------
    #pragma unroll
    for (int pass = 0; pass < 2; ++pass) {
        const v8f* acc = pass ? accB : accA;
        // bias + relu -> LDS
        #pragma unroll
        for (int t = 0; t < 4; ++t) {
            const int col  = t * 16 + mrow;
            const float bv = bfc1[col];
            #pragma unroll
            for (int r = 0; r < 8; ++r) {
                const int row = r + half * 8;
                h2s[(wave * 16 + row) * 68 + col] = fmaxf(acc[t][r] + bv, 0.f);
            }
        }
        __syncthreads();

        // 64->2 head + gumbel softmax straight-through; 1 thread per edge
        if (tid < 128) {
            const float* h2 = &h2s[tid * 68];
            float s0 = bfc2[0];
            float s1 = bfc2[1];
            #pragma unroll 8
            for (int c = 0; c < 64; ++c) {
                const float h = h2[c];
                s0 += h * Wfc2[c];
                s1 += h * Wfc2[64 + c];
            }
            // LDS row tid belongs to wave tid/16, tile row tid%16
            const long e2 = eBase + (long)(tid >> 4) * 32 + (pass ? 16 : 0)
                          + (tid & 15);
            const float g0 = gumbel[2 * e2];
            const float g1 = gumbel[2 * e2 + 1];
            const float z0 = (s0 + g0) * TAU_INV;
            const float z1 = (s1 + g1) * TAU_INV;
            const float mz  = fmaxf(z0, z1);
            const float ez0 = expf(z0 - mz);
            const float ez1 = expf(z1 - mz);
            const float y0  = ez0 / (ez0 + ez1);
            const float h0  = (z0 >= z1) ? 1.0f : 0.0f; // argmax -> one_hot[:,0]
            const float active = (h0 - y0) + y0;        // straight-through fwd
            out[e2]               = active;
            out[(long)NEDGE + e2] = 1.0f - active;
            out[2L * NEDGE + e2]  = 1.0f - active;
        }
        __syncthreads();   // protect h2s before tile-B overwrite / next pass
    }
}

// ------------------------------ launcher -----------------------------------
extern "C" void kernel_launch(void* const* d_in, const int* in_sizes, int n_in,
                              void* d_out, int out_size, void* d_ws, size_t ws_size,
                              hipStream_t stream) {
    const float* trace  = (const float*)d_in[0];
    const float* Wlin   = (const float*)d_in[1];
    const float* Wlin2  = (const float*)d_in[2];
    const float* Wfc1   = (const float*)d_in[3];
    const float* bfc1   = (const float*)d_in[4];
    const float* Wfc2   = (const float*)d_in[5];
    const float* bfc2   = (const float*)d_in[6];
    const float* gumbel = (const float*)d_in[7];
    const int*   eidx   = (const int*)d_in[8];
    // d_in[9] = num_edge (fixed 1.6M by construction; used as constant)

    float* mvc = (float*)d_ws;                 // N*128 floats = 51.2 MB
    float* out = (float*)d_out;                // 3*E floats

    node_mlp_kernel<<<NNODES / 16, 256, 0, stream>>>(trace, Wlin, Wlin2, mvc);
    edge_mask_kernel<<<NEDGE / 256, 256, 0, stream>>>(mvc, Wfc1, bfc1, Wfc2,
                                                      bfc2, gumbel, eidx, out);
    (void)in_sizes; (void)n_in; (void)out_size; (void)ws_size;
}